// OptimizedST_BIFNeuron_MS_15942918602770
// MI455X (gfx1250) — compile-verified
//
#include <hip/hip_runtime.h>

// ST-BIF neuron multi-step scan, fp32, streaming (zero-reuse) kernel.
// Per element i, per timestep t:
//   v += x[t];  pos = (v>=vth)&&(Tc<15);  neg = (v<0)&&(Tc>0)
//   spike = pos?1:(neg?-1:0);  v -= vth*spike;  Tc += spike;  out = vth*spike
// With vth=0.5 and PREFIRE=0 we track s = vth*spike and tq = vth*Tc:
//   pos: v>=0.5 && tq<7.5 ; neg: v<0 && tq>0 ; v-=s ; tq+=s ; out=s
// v0 = 0.5*vth = 0.25, tq0 = 0.

typedef float v4f __attribute__((ext_vector_type(4)));

#define BIF_VTH      0.5f
#define BIF_TMAX_Q   7.5f   /* (level-1)*vth = 15*0.5 */

__device__ __forceinline__ float bif_step(float xt, float& v, float& tq) {
    v += xt;
    const bool pos = (v >= BIF_VTH) && (tq < BIF_TMAX_Q);
    const bool neg = (v < 0.0f)     && (tq > 0.0f);
    const float s  = pos ? BIF_VTH : (neg ? -BIF_VTH : 0.0f);
    v  -= s;
    tq += s;
    return s;   // == spike * q_threshold
}

__global__ __launch_bounds__(256)
void bif_neuron_scan(const v4f* __restrict__ x,
                     const int* __restrict__ Tp,
                     v4f* __restrict__ out,
                     unsigned total4) {           // total elements / 4 (host-computed)
    const unsigned T      = (unsigned)Tp[0];      // scalar s_load broadcast
    const unsigned bf4    = total4 / T;           // float4 columns (32-bit udiv)
    const unsigned stride = gridDim.x * blockDim.x;
    unsigned       i      = blockIdx.x * blockDim.x + threadIdx.x;

    if (T == 8u) {
        // Fast path: issue all 8 strided B128 NT loads first (8 LOADcnt in
        // flight), then run the fully unrolled scan, NT-storing each step.
        for (; i < bf4; i += stride) {
            const unsigned nxt = i + stride;
            if (nxt < bf4)
                __builtin_prefetch(x + nxt, 0, 0);        // global_prefetch_b8

            v4f xs[8];
#pragma unroll
            for (unsigned t = 0; t < 8; ++t)
                xs[t] = __builtin_nontemporal_load(x + (t * bf4 + i));

            float v[4], tq[4];
#pragma unroll
            for (int e = 0; e < 4; ++e) { v[e] = 0.25f; tq[e] = 0.0f; }

#pragma unroll
            for (unsigned t = 0; t < 8; ++t) {
                v4f s;
#pragma unroll
                for (int e = 0; e < 4; ++e)
                    s[e] = bif_step(xs[t][e], v[e], tq[e]);
                __builtin_nontemporal_store(s, out + (t * bf4 + i));
            }
        }
    } else {
        // Generic T: per-timestep streamed load/compute/store.
        for (; i < bf4; i += stride) {
            float v[4], tq[4];
#pragma unroll
            for (int e = 0; e < 4; ++e) { v[e] = 0.25f; tq[e] = 0.0f; }

            for (unsigned t = 0; t < T; ++t) {
                const v4f xv = __builtin_nontemporal_load(x + (t * bf4 + i));
                v4f s;
#pragma unroll
                for (int e = 0; e < 4; ++e)
                    s[e] = bif_step(xv[e], v[e], tq[e]);
                __builtin_nontemporal_store(s, out + (t * bf4 + i));
            }
        }
    }
}

extern "C" void kernel_launch(void* const* d_in, const int* in_sizes, int n_in,
                              void* d_out, int out_size, void* d_ws, size_t ws_size,
                              hipStream_t stream) {
    const v4f* x   = (const v4f*)d_in[0];   // [T*B, C, H, W] fp32
    const int* Tp  = (const int*)d_in[1];   // scalar T (device, 1 element)
    v4f*       out = (v4f*)d_out;           // same shape/dtype as x

    const unsigned total  = (unsigned)in_sizes[0];
    const unsigned total4 = total / 4u;

    // Grid sized for the dominant T=8 case (one float4 column per thread,
    // single pass). The grid-stride loop keeps any other T correct.
    const int threads   = 256;                 // 8 wave32 waves
    unsigned  vec_items = total4 / 8u;         // bf4 at T=8
    if (vec_items < 1u) vec_items = 1u;
    unsigned blocks = (vec_items + threads - 1) / threads;

    bif_neuron_scan<<<blocks, threads, 0, stream>>>(x, Tp, out, total4);
}